// VectorQuantizer_62045097558120
// MI455X (gfx1250) — compile-verified
//
#include <hip/hip_runtime.h>

// VQ argmin + gather for MI455X (gfx1250), wave32 + WMMA f32 path.
//   score(row, code) = ||e||^2 - 2 * z.e     (||z||^2 dropped: constant per row)
// GEMM via V_WMMA_F32_16X16X4_F32 (exact fp32 — argmin is tie-sensitive).
// E chunks double-buffered in LDS; next chunk prefetched with
// GLOBAL_LOAD_ASYNC_TO_LDS_B128 (ASYNCcnt) while WMMAs run on current chunk.

#define N_EMBED         1024
#define EMBED_DIM       256
#define ZSTRIDE         260   // padded LDS row stride (floats): bank-conflict-free
#define ROWS_PER_WAVE   16
#define WAVES_PER_BLOCK 8
#define ROWS_PER_BLOCK  (ROWS_PER_WAVE * WAVES_PER_BLOCK)   // 128
#define CODES_PER_PASS  32    // two 16-code WMMA tiles per pass
#define NPASS           (N_EMBED / CODES_PER_PASS)          // 32

typedef __attribute__((ext_vector_type(2))) float v2f;
typedef __attribute__((ext_vector_type(8))) float v8f;
typedef __attribute__((ext_vector_type(4))) int   v4i;

#define AS_GLOBAL __attribute__((address_space(1)))
#define AS_LDS    __attribute__((address_space(3)))

#if defined(__gfx1250__) && __has_builtin(__builtin_amdgcn_global_load_async_to_lds_b128) && \
    __has_builtin(__builtin_amdgcn_s_wait_asynccnt)
#define USE_ASYNC 1
#else
#define USE_ASYNC 0
#endif

// ---------------- kernel 1: codebook row norms ||e||^2 -> d_ws ----------------
__global__ void vq_enorm_kernel(const float* __restrict__ emb,
                                float* __restrict__ enorm) {
  const int wv   = (blockIdx.x * blockDim.x + threadIdx.x) >> 5;  // one wave per code row
  const int lane = threadIdx.x & 31;
  if (wv >= N_EMBED) return;
  const float4* row = (const float4*)(emb + (size_t)wv * EMBED_DIM);
  float4 a = row[lane];
  float4 b = row[lane + 32];
  float s = a.x * a.x + a.y * a.y + a.z * a.z + a.w * a.w
          + b.x * b.x + b.y * b.y + b.z * b.z + b.w * b.w;
#pragma unroll
  for (int m = 16; m >= 1; m >>= 1) s += __shfl_xor(s, m, 32);
  if (lane == 0) enorm[wv] = s;
}

// Stage one 32-code chunk of E into (padded) LDS. Async path uses the CDNA5
// global->LDS async mover (per-lane 16B transfers, tracked by ASYNCcnt).
__device__ __forceinline__ void stage_chunk(const float* __restrict__ emb,
                                            float* Ebuf, int cb, int tid) {
#pragma unroll
  for (int j = 0; j < (CODES_PER_PASS * (EMBED_DIM / 4)) / 256; ++j) {  // 8 iters
    int i = tid + j * 256;
    int r = i >> 6;   // 64 float4 per code row
    int c = i & 63;
    const float* g = emb + (size_t)(cb + r) * EMBED_DIM + c * 4;
    float* l = Ebuf + r * ZSTRIDE + c * 4;
#if USE_ASYNC
    __builtin_amdgcn_global_load_async_to_lds_b128(
        (AS_GLOBAL v4i*)g, (AS_LDS v4i*)l, 0, 0);
#else
    *(float4*)l = *(const float4*)g;
#endif
  }
}

// ---------------- kernel 2: WMMA distance GEMM + argmin + gather ----------------
__global__ void __launch_bounds__(256)
vq_argmin_kernel(const float* __restrict__ z, const float* __restrict__ emb,
                 const float* __restrict__ enorm, float* __restrict__ zq,
                 float* __restrict__ idx_out) {
  extern __shared__ float smem[];
  float* Zs  = smem;                                             // 8*16*ZSTRIDE floats
  float* Es0 = Zs + WAVES_PER_BLOCK * ROWS_PER_WAVE * ZSTRIDE;   // 32*ZSTRIDE floats
  float* Es1 = Es0 + CODES_PER_PASS * ZSTRIDE;                   // 32*ZSTRIDE floats
  int* idxLds = (int*)(Es1 + CODES_PER_PASS * ZSTRIDE);          // 128 ints

  const int tid  = threadIdx.x;
  const int lane = tid & 31;
  const int w    = tid >> 5;
  const int blockRow0 = blockIdx.x * ROWS_PER_BLOCK;

  // kick off async prefetch of E chunk 0 while we stage Z synchronously
  stage_chunk(emb, Es0, 0, tid);

  // ---- stage the block's 128x256 Z tile into padded LDS ----
  {
    const float4* zg = (const float4*)(z + (size_t)blockRow0 * EMBED_DIM);
#pragma unroll 4
    for (int i = tid; i < ROWS_PER_BLOCK * (EMBED_DIM / 4); i += 256) {
      int r = i >> 6;
      int c = i & 63;
      *(float4*)&Zs[r * ZSTRIDE + c * 4] = zg[r * 64 + c];
    }
  }
#if USE_ASYNC
  __builtin_amdgcn_s_wait_asynccnt(0);
#endif
  __syncthreads();

  // per-lane running min/argmin; VGPR r of a C tile holds row (r + 8*(lane>=16))
  float minv[8];
  int   mini[8];
#pragma unroll
  for (int r = 0; r < 8; ++r) { minv[r] = 3.402823466e38f; mini[r] = 0; }

  const float* zw = Zs + w * ROWS_PER_WAVE * ZSTRIDE;
  // A (16x4 f32): lanes 0-15 -> M=lane, K=kk+{0,1}; lanes 16-31 -> M=lane-16, K=kk+{2,3}
  // B (4x16 f32): same per-lane pattern against each 16-code tile (B = E^T tile)
  const int frag_off = (lane & 15) * ZSTRIDE + ((lane >> 4) << 1);
  const int lcode    = lane & 15;

  for (int p = 0; p < NPASS; ++p) {
    const int cb = p * CODES_PER_PASS;
    float* cur = (p & 1) ? Es1 : Es0;
    float* nxt = (p & 1) ? Es0 : Es1;

    // prefetch next chunk into the other buffer (overlaps with WMMA below)
    if (p + 1 < NPASS) stage_chunk(emb, nxt, cb + CODES_PER_PASS, tid);

    v8f acc0 = {};   // 16 rows x codes [cb, cb+16)
    v8f acc1 = {};   // 16 rows x codes [cb+16, cb+32)
#pragma unroll 8
    for (int kk = 0; kk < EMBED_DIM; kk += 4) {
      v2f a  = *(const v2f*)&zw[frag_off + kk];                        // ds_load_b64
      v2f b0 = *(const v2f*)&cur[frag_off + kk];                       // ds_load_b64
      v2f b1 = *(const v2f*)&cur[frag_off + 16 * ZSTRIDE + kk];        // ds_load_b64
      acc0 = __builtin_amdgcn_wmma_f32_16x16x4_f32(false, a, false, b0,
                                                   (short)0, acc0, false, false);
      acc1 = __builtin_amdgcn_wmma_f32_16x16x4_f32(false, a, false, b1,
                                                   (short)0, acc1, false, false);
    }

    const float en0 = enorm[cb + lcode];
    const float en1 = enorm[cb + 16 + lcode];
    const int   c0  = cb + lcode;
    const int   c1  = cb + 16 + lcode;
#pragma unroll
    for (int r = 0; r < 8; ++r) {
      float s0 = fmaf(-2.0f, acc0[r], en0);   // strict '<' keeps first (smallest) index;
      bool b0c = s0 < minv[r];                // tile0 processed before tile1 (ascending codes)
      minv[r] = b0c ? s0 : minv[r];
      mini[r] = b0c ? c0 : mini[r];
      float s1 = fmaf(-2.0f, acc1[r], en1);
      bool b1c = s1 < minv[r];
      minv[r] = b1c ? s1 : minv[r];
      mini[r] = b1c ? c1 : mini[r];
    }

#if USE_ASYNC
    __builtin_amdgcn_s_wait_asynccnt(0);   // our prefetch writes landed in LDS
#endif
    __syncthreads();                       // everyone done reading cur / writing nxt
  }

  // butterfly reduce across the 16 lanes of each half (xor masks stay in-half)
#pragma unroll
  for (int r = 0; r < 8; ++r) {
#pragma unroll
    for (int m = 1; m <= 8; m <<= 1) {
      float ov = __shfl_xor(minv[r], m, 32);
      int   oi = __shfl_xor(mini[r], m, 32);
      if (ov < minv[r] || (ov == minv[r] && oi < mini[r])) {
        minv[r] = ov;
        mini[r] = oi;
      }
    }
  }

  if (lane == 0 || lane == 16) {
    const int half = lane >> 4;  // lanes 0-15 -> rows r; lanes 16-31 -> rows r+8
#pragma unroll
    for (int r = 0; r < 8; ++r) {
      int row = r + half * 8;
      idxLds[w * ROWS_PER_WAVE + row] = mini[r];
      // indices emitted as fp32 (output buffer is float-typed per harness)
      idx_out[blockRow0 + w * ROWS_PER_WAVE + row] = (float)mini[r];
    }
  }
  __syncthreads();

  // gather z_q = E[idx] for this wave's 16 rows (E rows are L2-resident)
#pragma unroll 2
  for (int r = 0; r < ROWS_PER_WAVE; ++r) {
    int code = idxLds[w * ROWS_PER_WAVE + r];
    const float4* src = (const float4*)(emb + (size_t)code * EMBED_DIM);
    int grow = blockRow0 + w * ROWS_PER_WAVE + r;
    float4* dst = (float4*)(zq + (size_t)grow * EMBED_DIM);
    dst[lane]      = src[lane];
    dst[lane + 32] = src[lane + 32];
  }
}

extern "C" void kernel_launch(void* const* d_in, const int* in_sizes, int n_in,
                              void* d_out, int out_size, void* d_ws, size_t ws_size,
                              hipStream_t stream) {
  const float* z   = (const float*)d_in[0];   // hidden_states (64,32,32,256) fp32
  const float* emb = (const float*)d_in[1];   // emb_weights   (1024,256)     fp32

  float* out   = (float*)d_out;
  float* zq    = out;                                   // 64*32*32*256 floats
  float* idx   = out + (size_t)64 * 32 * 32 * 256;      // 64*1024 index values
  float* enorm = (float*)d_ws;                          // 1024 floats of scratch

  const size_t lds_bytes =
      (size_t)(WAVES_PER_BLOCK * ROWS_PER_WAVE * ZSTRIDE +
               2 * CODES_PER_PASS * ZSTRIDE) * sizeof(float)
      + ROWS_PER_BLOCK * sizeof(int);  // ~200 KB, fits 320 KB WGP LDS
  (void)hipFuncSetAttribute((const void*)vq_argmin_kernel,
                            hipFuncAttributeMaxDynamicSharedMemorySize,
                            (int)lds_bytes);

  vq_enorm_kernel<<<dim3(N_EMBED / WAVES_PER_BLOCK), dim3(256), 0, stream>>>(emb, enorm);

  const int nrows = 64 * 32 * 32;  // 65536
  vq_argmin_kernel<<<dim3(nrows / ROWS_PER_BLOCK), dim3(256), lds_bytes, stream>>>(
      z, emb, enorm, zq, idx);
}